// PCHEB_59657095741758
// MI455X (gfx1250) — compile-verified
//
#include <hip/hip_runtime.h>

// ---------------------------------------------------------------------------
// ChebConv x3 GNN forward for MI455X (gfx1250, wave32).
//   N=50000 nodes, E=800000 edges, K=3, F: 128 -> 128 -> 128 -> 32
// Dense parts: V_WMMA_F32_16X16X4_F32 (exact f32 math). W is pre-transposed
//   in global memory, then staged per-workgroup into padded LDS rows via
//   gfx1250 async global->LDS b128 copies (ASYNCcnt); B-fragments are single
//   8-byte ds_load_b64 reads (no packing movs, <=2-way bank conflicts).
// Sparse propagation: wave-per-edge float4 gather + f32 global atomics.
// ---------------------------------------------------------------------------

#define USE_ASYNC_LDS 1

typedef __attribute__((ext_vector_type(2))) float v2f;
typedef __attribute__((ext_vector_type(8))) float v8f;

static constexpr int NN = 50000;
static constexpr int EE = 800000;
static constexpr int FF = 128;     // feature width of every prop input
static constexpr int HH = 128;
static constexpr int CC = 32;
static constexpr int LDPITCH = FF + 2;   // LDS row pitch (floats): 8B-aligned, conflict-light

// -------------------------------- utility ----------------------------------

__global__ void zero_kernel(float4* __restrict__ p, int n4) {
    int i = blockIdx.x * blockDim.x + threadIdx.x;
    if (i < n4) p[i] = make_float4(0.f, 0.f, 0.f, 0.f);
}

__global__ void deg_kernel(const int* __restrict__ src, float* __restrict__ deg, int e) {
    int i = blockIdx.x * blockDim.x + threadIdx.x;
    if (i < e) unsafeAtomicAdd(&deg[src[i]], 1.0f);
}

__global__ void dinv_kernel(const float* __restrict__ deg, float* __restrict__ dinv, int n) {
    int i = blockIdx.x * blockDim.x + threadIdx.x;
    if (i < n) {
        float d = deg[i];
        dinv[i] = (d > 0.f) ? rsqrtf(fmaxf(d, 1.f)) : 0.f;
    }
}

__global__ void edgew_kernel(const int* __restrict__ src, const int* __restrict__ dst,
                             const float* __restrict__ dinv, float* __restrict__ w, int e) {
    int i = blockIdx.x * blockDim.x + threadIdx.x;
    if (i < e) w[i] = -dinv[src[i]] * dinv[dst[i]];
}

// T2 = 2*T2 - T0   (elementwise, float4)
__global__ void cheb2_kernel(float4* __restrict__ t2, const float4* __restrict__ t0, int n4) {
    int i = blockIdx.x * blockDim.x + threadIdx.x;
    if (i < n4) {
        float4 a = t2[i];
        float4 b = t0[i];
        t2[i] = make_float4(2.f * a.x - b.x, 2.f * a.y - b.y,
                            2.f * a.z - b.z, 2.f * a.w - b.w);
    }
}

// W[3][128][FOUT] (row-major) -> Wt[3][FOUT][128]
__global__ void wtrans_kernel(const float* __restrict__ W, float* __restrict__ Wt, int fout) {
    int i = blockIdx.x * blockDim.x + threadIdx.x;
    int total = 3 * FF * fout;
    if (i >= total) return;
    int m = i / (FF * fout);
    int k = (i / fout) % FF;
    int c = i % fout;
    Wt[(size_t)m * fout * FF + (size_t)c * FF + k] = W[i];
}

// ------------------------- sparse propagation ------------------------------
// hout[dst] += w * hin[src]; one wave32 per edge, 4 floats per lane (=128).
__global__ __launch_bounds__(256)
void prop_kernel(const float* __restrict__ hin, const int* __restrict__ src,
                 const int* __restrict__ dst, const float* __restrict__ w,
                 float* __restrict__ hout, int e) {
    int gw   = (int)((blockIdx.x * blockDim.x + threadIdx.x) >> 5);
    int lane = threadIdx.x & 31;
    if (gw >= e) return;
    int   s  = src[gw];
    int   d  = dst[gw];
    float we = w[gw];
    float4 v = ((const float4*)(hin + (size_t)s * FF))[lane];
    float* op = hout + (size_t)d * FF + lane * 4;
    unsafeAtomicAdd(op + 0, we * v.x);
    unsafeAtomicAdd(op + 1, we * v.y);
    unsafeAtomicAdd(op + 2, we * v.z);
    unsafeAtomicAdd(op + 3, we * v.w);
}

// --------------------------- fused WMMA GEMM -------------------------------
// out[r, c] = act( sum_m T_m[r, :] @ W[m, :, c] + bias[c] )
// Wt layout: [3][FOUT][128] (transposed weights).
// Block = 8 waves; each wave owns one 16-row stripe and NT = FOUT/16 tiles.
// Wt is staged one m-slice at a time into padded LDS rows (pitch 130 floats)
// via async global->LDS b128 copies; B-fragments are single ds_load_b64.
// ACT: 0 = none, 1 = relu, 2 = tanh
template <int FOUT, int ACT>
__global__ __launch_bounds__(256)
void cheb_gemm_kernel(const float* __restrict__ T0, const float* __restrict__ T1,
                      const float* __restrict__ T2, const float* __restrict__ Wt,
                      const float* __restrict__ bias, float* __restrict__ out,
                      int ntiles) {
    constexpr int NT = FOUT / 16;
    __shared__ float sWt[FOUT * LDPITCH];    // one m-slice of Wt, padded rows

    const int wid  = threadIdx.x >> 5;
    const int tile = blockIdx.x * 8 + wid;
    const bool active = (tile < ntiles);     // wave-uniform

    const int lane = threadIdx.x & 31;
    const int half = lane >> 4;              // 0: K={0,1}  1: K={2,3}
    const int l16  = lane & 15;
    const size_t row = (size_t)(active ? tile : 0) * 16 + l16;

    v8f acc[NT];
#pragma unroll
    for (int t = 0; t < NT; ++t)
#pragma unroll
        for (int j = 0; j < 8; ++j) acc[t][j] = 0.f;

    const float* Ts[3] = {T0, T1, T2};
#pragma unroll
    for (int m = 0; m < 3; ++m) {
        __syncthreads();                     // previous slice fully consumed

        // ---- stage Wt[m] into LDS (padded rows): 16B per lane per chunk ----
        {
            const float* gsrc = Wt + (size_t)m * FOUT * FF;
            constexpr int CHUNKS = (FOUT * FF) / (256 * 4);   // float4s per thread
#pragma unroll
            for (int i = 0; i < CHUNKS; ++i) {
                int idx4 = i * 256 + threadIdx.x;
                int c    = idx4 >> 5;        // FF/4 = 32 chunks per row
                int k4   = idx4 & 31;
                float* ldst = &sWt[c * LDPITCH + k4 * 4];
#if USE_ASYNC_LDS
                unsigned lds_off = (unsigned)(size_t)(void*)ldst;
                unsigned long long ga =
                    (unsigned long long)(size_t)(gsrc + (size_t)idx4 * 4);
                asm volatile("global_load_async_to_lds_b128 %0, %1, off"
                             :: "v"(lds_off), "v"(ga) : "memory");
#else
                *(float4*)ldst = *(const float4*)(gsrc + (size_t)idx4 * 4);
#endif
            }
#if USE_ASYNC_LDS
            asm volatile("s_wait_asynccnt 0" ::: "memory");
#endif
        }
        __syncthreads();                     // slice visible to all waves

        if (active) {
            const float* Trow = Ts[m] + row * FF;
            for (int k = 0; k < FF; k += 4) {
                // A fragment (16x4 f32): lanes 0-15 K={k,k+1}; 16-31 K={k+2,k+3}
                v2f a = *(const v2f*)(Trow + k + half * 2);
#pragma unroll
                for (int t = 0; t < NT; ++t) {
                    // B fragment: contiguous 8B in transposed LDS -> ds_load_b64
                    v2f b = *(const v2f*)&sWt[(t * 16 + l16) * LDPITCH + k + half * 2];
                    acc[t] = __builtin_amdgcn_wmma_f32_16x16x4_f32(
                        false, a, false, b, (short)0, acc[t], false, false);
                }
            }
        }
    }

    if (active) {
        // C/D layout: VGPR j -> M = j + half*8, N = l16 (per tile)
#pragma unroll
        for (int t = 0; t < NT; ++t) {
            int c = t * 16 + l16;
            float bv = bias[c];
#pragma unroll
            for (int j = 0; j < 8; ++j) {
                size_t r = (size_t)tile * 16 + half * 8 + j;
                float v = acc[t][j] + bv;
                if (ACT == 1) v = fmaxf(v, 0.f);
                if (ACT == 2) v = tanhf(v);
                out[r * FOUT + c] = v;
            }
        }
    }
}

// ------------------------------ orchestration ------------------------------

static inline void zero_buf(float* p, size_t nfloats, hipStream_t s) {
    int n4 = (int)(nfloats / 4);
    zero_kernel<<<(n4 + 255) / 256, 256, 0, s>>>((float4*)p, n4);
}

extern "C" void kernel_launch(void* const* d_in, const int* in_sizes, int n_in,
                              void* d_out, int out_size, void* d_ws, size_t ws_size,
                              hipStream_t stream) {
    const float* x  = (const float*)d_in[0];
    const int*   ei = (const int*)d_in[1];
    const float* W1 = (const float*)d_in[2];
    const float* b1 = (const float*)d_in[3];
    const float* W2 = (const float*)d_in[4];
    const float* b2 = (const float*)d_in[5];
    const float* W3 = (const float*)d_in[6];
    const float* b3 = (const float*)d_in[7];
    float* out = (float*)d_out;

    const int* src = ei;        // edge_index[0]
    const int* dst = ei + EE;   // edge_index[1]

    float* ws   = (float*)d_ws;
    float* deg  = ws;  ws += NN;
    float* dinv = ws;  ws += NN;
    float* wbuf = ws;  ws += EE;
    float* wt1  = ws;  ws += 3 * FF * HH;       // W1^T
    float* wt2  = ws;  ws += 3 * FF * HH;       // W2^T
    float* wt3  = ws;  ws += 3 * FF * CC;       // W3^T
    float* bufA = ws;  ws += (size_t)NN * FF;   // T1
    float* bufB = ws;  ws += (size_t)NN * FF;   // T2
    float* bufC = ws;  ws += (size_t)NN * FF;   // h1
    float* bufD = ws;  ws += (size_t)NN * FF;   // h2

    const int TB = 256;
    dim3 blk(TB);
    dim3 grid_e((EE + TB - 1) / TB);
    dim3 grid_n((NN + TB - 1) / TB);
    dim3 grid_prop(EE / 8);                       // one wave per edge
    dim3 grid_ew((NN * FF / 4 + TB - 1) / TB);    // elementwise float4
    const int ntiles = NN / 16;                   // 3125
    dim3 grid_gemm((ntiles + 7) / 8);

    // --- graph normalization (deg, dinv, edge weights) + weight transposes ---
    zero_buf(deg, NN, stream);
    deg_kernel<<<grid_e, blk, 0, stream>>>(src, deg, EE);
    dinv_kernel<<<grid_n, blk, 0, stream>>>(deg, dinv, NN);
    edgew_kernel<<<grid_e, blk, 0, stream>>>(src, dst, dinv, wbuf, EE);
    wtrans_kernel<<<(3 * FF * HH + TB - 1) / TB, blk, 0, stream>>>(W1, wt1, HH);
    wtrans_kernel<<<(3 * FF * HH + TB - 1) / TB, blk, 0, stream>>>(W2, wt2, HH);
    wtrans_kernel<<<(3 * FF * CC + TB - 1) / TB, blk, 0, stream>>>(W3, wt3, CC);

    // --- layer 1: x -> h1 (relu) ---
    zero_buf(bufA, (size_t)NN * FF, stream);
    prop_kernel<<<grid_prop, blk, 0, stream>>>(x, src, dst, wbuf, bufA, EE);
    zero_buf(bufB, (size_t)NN * FF, stream);
    prop_kernel<<<grid_prop, blk, 0, stream>>>(bufA, src, dst, wbuf, bufB, EE);
    cheb2_kernel<<<grid_ew, blk, 0, stream>>>((float4*)bufB, (const float4*)x, NN * FF / 4);
    cheb_gemm_kernel<HH, 1><<<grid_gemm, blk, 0, stream>>>(x, bufA, bufB, wt1, b1, bufC, ntiles);

    // --- layer 2: h1 -> h2 (relu) ---
    zero_buf(bufA, (size_t)NN * FF, stream);
    prop_kernel<<<grid_prop, blk, 0, stream>>>(bufC, src, dst, wbuf, bufA, EE);
    zero_buf(bufB, (size_t)NN * FF, stream);
    prop_kernel<<<grid_prop, blk, 0, stream>>>(bufA, src, dst, wbuf, bufB, EE);
    cheb2_kernel<<<grid_ew, blk, 0, stream>>>((float4*)bufB, (const float4*)bufC, NN * FF / 4);
    cheb_gemm_kernel<HH, 1><<<grid_gemm, blk, 0, stream>>>(bufC, bufA, bufB, wt2, b2, bufD, ntiles);

    // --- layer 3: h2 -> out (tanh) ---
    zero_buf(bufA, (size_t)NN * FF, stream);
    prop_kernel<<<grid_prop, blk, 0, stream>>>(bufD, src, dst, wbuf, bufA, EE);
    zero_buf(bufB, (size_t)NN * FF, stream);
    prop_kernel<<<grid_prop, blk, 0, stream>>>(bufA, src, dst, wbuf, bufB, EE);
    cheb2_kernel<<<grid_ew, blk, 0, stream>>>((float4*)bufB, (const float4*)bufD, NN * FF / 4);
    cheb_gemm_kernel<CC, 2><<<grid_gemm, blk, 0, stream>>>(bufD, bufA, bufB, wt3, b3, out, ntiles);
}